// TopoGNN_4724464025664
// MI455X (gfx1250) — compile-verified
//
#include <hip/hip_runtime.h>

// ---------------- constants (match reference) ----------------
#define NN 100000
#define EE 500000
#define GG 128
#define NDIM 91
#define EDIM 20
#define HH 128
#define LL 3
#define EPS 1e-5f
#define K1P 276          // 128 + 128 + 20 (folded edge-feature K), multiple of 4
#define KNP 92           // 91 padded to multiple of 4

typedef float v2f __attribute__((ext_vector_type(2)));
typedef float v8f __attribute__((ext_vector_type(8)));

static __device__ __forceinline__ v8f wmma_f32_k4(v2f a, v2f b, v8f c) {
  // V_WMMA_F32_16X16X4_F32 : D = A(16x4) * B(4x16) + C(16x16), fp32
  return __builtin_amdgcn_wmma_f32_16x16x4_f32(
      false, a, false, b, (short)0, c, false, false);
}

// B fragment from pair-packed weights: Wp[(k>>1)*256 + n*2 + (k&1)]
static __device__ __forceinline__ v2f bfrag(const float* __restrict__ Wp,
                                            int kb, int n) {
  return *(const v2f*)&Wp[(size_t)(kb >> 1) * (2 * HH) + n * 2];
}

// monotonic float<->uint mapping for atomic max with negatives
static __device__ __forceinline__ unsigned fmono(float f) {
  unsigned u = __float_as_uint(f);
  return (u & 0x80000000u) ? ~u : (u | 0x80000000u);
}
static __device__ __forceinline__ float funmono(unsigned u) {
  return (u & 0x80000000u) ? __uint_as_float(u & 0x7FFFFFFFu)
                           : __uint_as_float(~u);
}

// ---------------- zero fill ----------------
__global__ void k_zero(float* __restrict__ p, long n) {
  long i = (long)blockIdx.x * blockDim.x + threadIdx.x;
  long stride = (long)gridDim.x * blockDim.x;
  for (; i < n; i += stride) p[i] = 0.f;
}

// ---------------- weight pre-pass kernels (tiny, once per launch) -----------
// pack [K][128] row-major -> pair layout, zero-padded to Kp rows
__global__ void k_pack_pairs(const float* __restrict__ src,
                             float* __restrict__ dst, int K, int Kp) {
  int idx = blockIdx.x * blockDim.x + threadIdx.x;
  if (idx >= Kp * HH) return;
  int k = idx / HH, n = idx % HH;
  float v = (k < K) ? src[(size_t)k * HH + n] : 0.f;
  dst[(size_t)(k >> 1) * (2 * HH) + n * 2 + (k & 1)] = v;
}

// fold edge-encoder into msg GEMM1: Wc[20,128] = eeW @ W1c ; b1p = b1 + eeb @ W1c
__global__ void k_fold_edge(const float* __restrict__ eeW,
                            const float* __restrict__ W1c,
                            const float* __restrict__ eeb,
                            const float* __restrict__ b1,
                            float* __restrict__ Wc, float* __restrict__ b1p) {
  int idx = blockIdx.x * blockDim.x + threadIdx.x;
  if (idx < EDIM * HH) {
    int k = idx / HH, n = idx % HH;
    float acc = 0.f;
    for (int j = 0; j < HH; ++j)
      acc += eeW[(size_t)k * HH + j] * W1c[(size_t)j * HH + n];
    Wc[idx] = acc;
  } else if (idx < EDIM * HH + HH) {
    int n = idx - EDIM * HH;
    float acc = b1[n];
    for (int j = 0; j < HH; ++j) acc += eeb[j] * W1c[(size_t)j * HH + n];
    b1p[n] = acc;
  }
}

// pack msg W1': rows 0..255 = W1[0:256], rows 256..275 = Wc (folded edge part)
__global__ void k_pack_msgw1(const float* __restrict__ W1,
                             const float* __restrict__ Wc,
                             float* __restrict__ dst) {
  int idx = blockIdx.x * blockDim.x + threadIdx.x;
  if (idx >= K1P * HH) return;
  int k = idx / HH, n = idx % HH;
  float v = (k < 2 * HH) ? W1[(size_t)k * HH + n]
                         : Wc[(size_t)(k - 2 * HH) * HH + n];
  dst[(size_t)(k >> 1) * (2 * HH) + n * 2 + (k & 1)] = v;
}

// ---------------- node encoder: h = node_feats @ ne_W + ne_b ----------------
__global__ void __launch_bounds__(32)
k_node_encode(const float* __restrict__ nf, const float* __restrict__ Wp,
              const float* __restrict__ b, float* __restrict__ h) {
  __shared__ __align__(16) float sA[16][KNP];
  const int m0 = blockIdx.x * 16;
  const int t = threadIdx.x;
  for (int idx = t; idx < 16 * NDIM; idx += 32) {
    int r = idx / NDIM, c = idx % NDIM;
    sA[r][c] = nf[(size_t)(m0 + r) * NDIM + c];
  }
  if (t < 16) sA[t][NDIM] = 0.f;
  __syncthreads();
  const int ln = t & 15, hi = t >> 4;
  for (int nt = 0; nt < 8; ++nt) {
    const int n0 = nt * 16;
    v8f c;
    const float bias = b[n0 + ln];
#pragma unroll
    for (int r = 0; r < 8; ++r) c[r] = bias;
#pragma unroll 2
    for (int k = 0; k < KNP; k += 4) {
      const int kb = k + 2 * hi;
      v2f a = *(const v2f*)&sA[ln][kb];
      c = wmma_f32_k4(a, bfrag(Wp, kb, n0 + ln), c);
    }
#pragma unroll
    for (int r = 0; r < 8; ++r)
      h[(size_t)(m0 + r + 8 * hi) * HH + n0 + ln] = c[r];
  }
}

// ---------------- message kernel (one layer) ----------------
// one wave per 32-edge tile (2 x 16-row subtiles share every B fragment):
//   m_in = [h[src] | h[dst] | ef]   (edge encoder folded into W1p/b1p)
//   msg  = relu(m_in @ W1p + b1p) @ W2p + b2 ; atomic scatter into agg[dst]
__global__ void __launch_bounds__(32)
k_msg(const float* __restrict__ h, const int* __restrict__ ei,
      const float* __restrict__ ef, const float* __restrict__ W1p,
      const float* __restrict__ b1p, const float* __restrict__ W2p,
      const float* __restrict__ b2, float* __restrict__ agg) {
  __shared__ __align__(16) float sM[32][K1P];    // 34.5 KB
  __shared__ __align__(16) float sMid[32][HH];   // 16 KB
  __shared__ int sIdx[64];                        // [0..31]=src, [32..63]=dst
  const int e0 = blockIdx.x * 32;
  const int t = threadIdx.x;
  sIdx[t] = ei[e0 + t];
  sIdx[32 + t] = ei[EE + e0 + t];
  __syncthreads();
  {  // gather: lane t owns row t
    const float* srow = h + (size_t)sIdx[t] * HH;
    const float* drow = h + (size_t)sIdx[32 + t] * HH;
#pragma unroll
    for (int q = 0; q < HH; q += 4) {
      *(float4*)&sM[t][q] = *(const float4*)&srow[q];
      *(float4*)&sM[t][HH + q] = *(const float4*)&drow[q];
    }
  }
  for (int idx = t; idx < 32 * EDIM; idx += 32) {
    int r = idx / EDIM, c = idx % EDIM;
    sM[r][2 * HH + c] = ef[(size_t)(e0 + r) * EDIM + c];
  }
  __syncthreads();
  const int ln = t & 15, hi = t >> 4;
  // --- GEMM1: [32,276] x [276,128] + b1p, relu -> sMid ---
  for (int nt = 0; nt < 8; ++nt) {
    const int n0 = nt * 16;
    v8f c0, c1;
    const float bias = b1p[n0 + ln];
#pragma unroll
    for (int r = 0; r < 8; ++r) { c0[r] = bias; c1[r] = bias; }
    __builtin_prefetch(&W1p[(size_t)hi * (2 * HH) + (n0 + ln) * 2], 0, 3);
#pragma unroll 2
    for (int k = 0; k < K1P; k += 4) {
      const int kb = k + 2 * hi;
      v2f bb = bfrag(W1p, kb, n0 + ln);
      v2f a0 = *(const v2f*)&sM[ln][kb];
      v2f a1 = *(const v2f*)&sM[16 + ln][kb];
      c0 = wmma_f32_k4(a0, bb, c0);
      c1 = wmma_f32_k4(a1, bb, c1);
    }
#pragma unroll
    for (int r = 0; r < 8; ++r) {
      const int M = r + 8 * hi;
      sMid[M][n0 + ln] = fmaxf(c0[r], 0.f);
      sMid[16 + M][n0 + ln] = fmaxf(c1[r], 0.f);
    }
  }
  __syncthreads();
  // --- GEMM2: [32,128] x [128,128] + b2 -> scatter into agg[dst] ---
  for (int nt = 0; nt < 8; ++nt) {
    const int n0 = nt * 16;
    v8f c0, c1;
    const float bias = b2[n0 + ln];
#pragma unroll
    for (int r = 0; r < 8; ++r) { c0[r] = bias; c1[r] = bias; }
#pragma unroll 2
    for (int k = 0; k < HH; k += 4) {
      const int kb = k + 2 * hi;
      v2f bb = bfrag(W2p, kb, n0 + ln);
      v2f a0 = *(const v2f*)&sMid[ln][kb];
      v2f a1 = *(const v2f*)&sMid[16 + ln][kb];
      c0 = wmma_f32_k4(a0, bb, c0);
      c1 = wmma_f32_k4(a1, bb, c1);
    }
#pragma unroll
    for (int r = 0; r < 8; ++r) {
      const int M = r + 8 * hi;
      atomicAdd(&agg[(size_t)sIdx[32 + M] * HH + n0 + ln], c0[r]);
      atomicAdd(&agg[(size_t)sIdx[32 + 16 + M] * HH + n0 + ln], c1[r]);
    }
  }
}

// ---------------- node update + layernorm (one layer) ----------------
// h = LN(h + relu([h|agg] @ upd_W + upd_b)) ; in-place (tile-local rows)
__global__ void __launch_bounds__(32)
k_update(float* __restrict__ h, const float* __restrict__ agg,
         const float* __restrict__ Wp, const float* __restrict__ b,
         const float* __restrict__ lg, const float* __restrict__ lb) {
  __shared__ __align__(16) float sC[16][2 * HH];
  const int m0 = blockIdx.x * 16;
  const int t = threadIdx.x;
  {
    const int m = t >> 1, half = t & 1;
    const float* hr = h + (size_t)(m0 + m) * HH + half * 64;
    const float* ar = agg + (size_t)(m0 + m) * HH + half * 64;
    float* o0 = &sC[m][half * 64];
    float* o1 = &sC[m][HH + half * 64];
#pragma unroll
    for (int q = 0; q < 64; q += 4) {
      *(float4*)&o0[q] = *(const float4*)&hr[q];
      *(float4*)&o1[q] = *(const float4*)&ar[q];
    }
  }
  __syncthreads();
  const int ln = t & 15, hi = t >> 4;
  float x[8][8];
  for (int nt = 0; nt < 8; ++nt) {
    const int n0 = nt * 16;
    v8f c;
    const float bias = b[n0 + ln];
#pragma unroll
    for (int r = 0; r < 8; ++r) c[r] = bias;
#pragma unroll 2
    for (int k = 0; k < 2 * HH; k += 4) {
      const int kb = k + 2 * hi;
      v2f a = *(const v2f*)&sC[ln][kb];
      c = wmma_f32_k4(a, bfrag(Wp, kb, n0 + ln), c);
    }
#pragma unroll
    for (int r = 0; r < 8; ++r)
      x[nt][r] = sC[r + 8 * hi][n0 + ln] + fmaxf(c[r], 0.f);
  }
  // layernorm over 128 features of row M = r + 8*hi
  float s1[8], s2[8];
#pragma unroll
  for (int r = 0; r < 8; ++r) { s1[r] = 0.f; s2[r] = 0.f; }
#pragma unroll
  for (int nt = 0; nt < 8; ++nt)
#pragma unroll
    for (int r = 0; r < 8; ++r) {
      s1[r] += x[nt][r];
      s2[r] += x[nt][r] * x[nt][r];
    }
#pragma unroll
  for (int mask = 1; mask <= 8; mask <<= 1)
#pragma unroll
    for (int r = 0; r < 8; ++r) {
      s1[r] += __shfl_xor(s1[r], mask, 32);
      s2[r] += __shfl_xor(s2[r], mask, 32);
    }
  float mu[8], rs[8];
#pragma unroll
  for (int r = 0; r < 8; ++r) {
    mu[r] = s1[r] * (1.f / HH);
    float var = s2[r] * (1.f / HH) - mu[r] * mu[r];
    rs[r] = rsqrtf(var + EPS);
  }
  for (int nt = 0; nt < 8; ++nt) {
    const int n0 = nt * 16;
    const float gv = lg[n0 + ln];
    const float bv = lb[n0 + ln];
#pragma unroll
    for (int r = 0; r < 8; ++r)
      h[(size_t)(m0 + r + 8 * hi) * HH + n0 + ln] =
          (x[nt][r] - mu[r]) * rs[r] * gv + bv;
  }
}

// ---------------- graph pooling ----------------
__global__ void __launch_bounds__(128)
k_pool(const float* __restrict__ h, const int* __restrict__ batch,
       float* __restrict__ psum, unsigned* __restrict__ pmax,
       float* __restrict__ pcnt) {
  const int node = blockIdx.x;
  const int n = threadIdx.x;
  const float v = h[(size_t)node * HH + n];
  const int g = batch[node];
  atomicAdd(&psum[(size_t)g * HH + n], v);
  atomicMax(&pmax[(size_t)g * HH + n], fmono(v));
  if (n == 0) atomicAdd(&pcnt[g], 1.f);
}

// ---------------- readout MLP: [G,256] -> 128 -> 64 -> 1 ----------------
__global__ void __launch_bounds__(128)
k_readout(const float* __restrict__ psum, const unsigned* __restrict__ pmax,
          const float* __restrict__ pcnt, const float* __restrict__ W1,
          const float* __restrict__ b1, const float* __restrict__ W2,
          const float* __restrict__ b2, const float* __restrict__ W3,
          const float* __restrict__ b3, float* __restrict__ out) {
  __shared__ float hg[2 * HH];
  __shared__ float x1[HH];
  __shared__ float x2[HH / 2];
  const int g = blockIdx.x;
  const int n = threadIdx.x;
  const float denom = pcnt[g] + 1.f;  // torch include_self mean
  hg[n] = psum[(size_t)g * HH + n] / denom;
  hg[HH + n] = funmono(pmax[(size_t)g * HH + n]);
  __syncthreads();
  {
    float acc = b1[n];
    for (int k = 0; k < 2 * HH; ++k) acc += hg[k] * W1[(size_t)k * HH + n];
    x1[n] = fmaxf(acc, 0.f);
  }
  __syncthreads();
  if (n < HH / 2) {
    float acc = b2[n];
    for (int k = 0; k < HH; ++k) acc += x1[k] * W2[(size_t)k * (HH / 2) + n];
    x2[n] = fmaxf(acc, 0.f);
  }
  __syncthreads();
  if (n == 0) {
    float acc = b3[0];
    for (int k = 0; k < HH / 2; ++k) acc += x2[k] * W3[k];
    out[g] = acc;
  }
}

// ---------------- host launcher ----------------
extern "C" void kernel_launch(void* const* d_in, const int* in_sizes, int n_in,
                              void* d_out, int out_size, void* d_ws,
                              size_t ws_size, hipStream_t stream) {
  const float* node_feats = (const float*)d_in[0];
  const int*   edge_index = (const int*)d_in[1];
  const float* edge_feats = (const float*)d_in[2];
  const int*   batch      = (const int*)d_in[3];
  const float* ne_W = (const float*)d_in[4];
  const float* ne_b = (const float*)d_in[5];
  const float* ee_W = (const float*)d_in[6];
  const float* ee_b = (const float*)d_in[7];
  const float* msg_W1 = (const float*)d_in[8];
  const float* msg_b1 = (const float*)d_in[9];
  const float* msg_W2 = (const float*)d_in[10];
  const float* msg_b2 = (const float*)d_in[11];
  const float* upd_W  = (const float*)d_in[12];
  const float* upd_b  = (const float*)d_in[13];
  const float* ln_g   = (const float*)d_in[14];
  const float* ln_b   = (const float*)d_in[15];
  const float* ro_W1 = (const float*)d_in[16];
  const float* ro_b1 = (const float*)d_in[17];
  const float* ro_W2 = (const float*)d_in[18];
  const float* ro_b2 = (const float*)d_in[19];
  const float* ro_W3 = (const float*)d_in[20];
  const float* ro_b3 = (const float*)d_in[21];
  float* out = (float*)d_out;

  // workspace layout (floats)
  float* p = (float*)d_ws;
  float* h    = p;  p += (size_t)NN * HH;
  float* agg  = p;  p += (size_t)NN * HH;
  float* psum = p;  p += (size_t)GG * HH;
  unsigned* pmax = (unsigned*)p;  p += (size_t)GG * HH;
  float* pcnt = p;  p += GG;
  float* neWp = p;  p += (size_t)KNP * HH;
  float* w1p  = p;  p += (size_t)LL * K1P * HH;
  float* w2p  = p;  p += (size_t)LL * HH * HH;
  float* updp = p;  p += (size_t)LL * 2 * HH * HH;
  float* b1p  = p;  p += (size_t)LL * HH;
  float* wc   = p;  p += (size_t)EDIM * HH;

  // ---- weight pre-pass (tiny) ----
  k_pack_pairs<<<(KNP * HH + 255) / 256, 256, 0, stream>>>(ne_W, neWp, NDIM, KNP);
  for (int i = 0; i < LL; ++i) {
    const float* W1i = msg_W1 + (size_t)i * 3 * HH * HH;
    k_fold_edge<<<(EDIM * HH + HH + 255) / 256, 256, 0, stream>>>(
        ee_W, W1i + (size_t)2 * HH * HH, ee_b, msg_b1 + (size_t)i * HH,
        wc, b1p + (size_t)i * HH);
    k_pack_msgw1<<<(K1P * HH + 255) / 256, 256, 0, stream>>>(
        W1i, wc, w1p + (size_t)i * K1P * HH);
    k_pack_pairs<<<(HH * HH + 255) / 256, 256, 0, stream>>>(
        msg_W2 + (size_t)i * HH * HH, w2p + (size_t)i * HH * HH, HH, HH);
    k_pack_pairs<<<(2 * HH * HH + 255) / 256, 256, 0, stream>>>(
        upd_W + (size_t)i * 2 * HH * HH, updp + (size_t)i * 2 * HH * HH,
        2 * HH, 2 * HH);
  }

  // ---- encode nodes ----
  k_node_encode<<<NN / 16, 32, 0, stream>>>(node_feats, neWp, ne_b, h);

  // ---- message-passing layers ----
  for (int i = 0; i < LL; ++i) {
    k_zero<<<2048, 256, 0, stream>>>(agg, (long)NN * HH);
    k_msg<<<EE / 32, 32, 0, stream>>>(
        h, edge_index, edge_feats, w1p + (size_t)i * K1P * HH,
        b1p + (size_t)i * HH, w2p + (size_t)i * HH * HH,
        msg_b2 + (size_t)i * HH, agg);
    k_update<<<NN / 16, 32, 0, stream>>>(
        h, agg, updp + (size_t)i * 2 * HH * HH, upd_b + (size_t)i * HH,
        ln_g + (size_t)i * HH, ln_b + (size_t)i * HH);
  }

  // ---- pooling (0 < fmono(-inf), so zero-init is a valid max identity) ----
  k_zero<<<64, 256, 0, stream>>>(psum, (long)(2 * GG * HH + GG));
  k_pool<<<NN, HH, 0, stream>>>(h, batch, psum, pmax, pcnt);

  // ---- readout ----
  k_readout<<<GG, HH, 0, stream>>>(psum, pmax, pcnt, ro_W1, ro_b1, ro_W2,
                                   ro_b2, ro_W3, ro_b3, out);
}